// FCOS3DPostProcessor_11630771437842
// MI455X (gfx1250) — compile-verified
//
#include <hip/hip_runtime.h>
#include <hip/hip_bf16.h>

#define H_BEV 350
#define W_BEV 300
#define HW (H_BEV * W_BEV)        // 105000
#define ACH 6                     // NUM_ANGLES * NUM_CLASSES
#define NCLS 3
#define HWAC (HW * ACH)           // 630000
#define K_TOP 256
#define REG_DIM 24

typedef float v2f __attribute__((ext_vector_type(2)));
typedef float v8f __attribute__((ext_vector_type(8)));
typedef unsigned int u32x4 __attribute__((ext_vector_type(4)));
typedef int i32x4 __attribute__((ext_vector_type(4)));
typedef int i32x8 __attribute__((ext_vector_type(8)));

__device__ __forceinline__ float sigmoidf(float x) { return 1.0f / (1.0f + __expf(-x)); }

// ---------------- Kernel 1: TDM tile -> LDS, sigmoid scores + candidate mask ----------------
// Iterates in raw box_cls memory order (img, ch, hw) so each block consumes one
// contiguous 256-float tile, DMA'd into LDS by the Tensor Data Mover.
__global__ void score_kernel(const float* __restrict__ cls,
                             const float* __restrict__ ctr,
                             float* __restrict__ scores, int total) {
    __shared__ float tile[256];
    int base = blockIdx.x * 256;
    int g = base + threadIdx.x;

    if (threadIdx.x < 32) {   // wave 0 issues the tensor DMA (EXEC ignored by TDM; branch skips other waves)
        unsigned long long ga = (unsigned long long)(const void*)(cls + base);
        unsigned int lds_off = (unsigned int)(unsigned long long)(void*)&tile[0];
        unsigned int dim0 = (unsigned int)(total - base);   // OOB tail reads return zero

        u32x4 g0;
        g0[0] = 1u;                                          // count=1, user mode, no gather
        g0[1] = lds_off;                                     // lds_addr (bytes)
        g0[2] = (unsigned int)(ga & 0xFFFFFFFFull);          // global_addr[31:0]
        g0[3] = (unsigned int)((ga >> 32) & 0x1FFFFFFull)    // global_addr[56:32]
                | (2u << 30);                                // type = 2 ("image")

        i32x8 g1;
        g1[0] = (int)(2u << 16);                             // data_size = 4 bytes
        g1[1] = (int)((dim0 & 0xFFFFu) << 16);               // tensor_dim0[15:0]
        g1[2] = (int)(((dim0 >> 16) & 0xFFFFu)               // tensor_dim0[31:16]
                | (1u << 16));                               // tensor_dim1 = 1
        g1[3] = (int)(256u << 16);                           // tile_dim0 = 256
        g1[4] = 0;                                           // tile_dim1 = tile_dim2 = 0 (unused)
        g1[5] = (int)dim0;                                   // tensor_dim0_stride[31:0]
        g1[6] = 0;
        g1[7] = 0;

        i32x4 z4 = {};
#if __clang_major__ >= 23
        i32x8 z8 = {};
        __builtin_amdgcn_tensor_load_to_lds(g0, g1, z4, z4, z8, 0);
#else
        __builtin_amdgcn_tensor_load_to_lds(g0, g1, z4, z4, 0);
#endif
        __builtin_amdgcn_s_wait_tensorcnt(0);
    }
    __syncthreads();

    if (g < total) {
        int img = g / HWAC;
        int rr = g - img * HWAC;
        int ch = rr / HW;
        int hw = rr - ch * HW;
        int a = ch / NCLS;
        float sc = sigmoidf(tile[threadIdx.x]);
        const float* ct = ctr + (size_t)img * (2 * HW) + (size_t)a * HW + hw;
        __builtin_prefetch(ct + 256, 0, 1);                  // global_prefetch_b8
        float sct = sigmoidf(*ct);
        int r = hw * ACH + ch;                               // flat (HW, A, C) score index
        scores[(size_t)img * HWAC + r] = (sc > 0.05f) ? sc * sct : -1.0f;
    }
}

// ---------------- Kernel 2: exact top-256 per image ----------------
__global__ void topk_kernel(float* __restrict__ scores,
                            int* __restrict__ topidx, float* __restrict__ topval) {
    int img = blockIdx.x;
    int t = threadIdx.x;
    float* s = scores + (size_t)img * HWAC;
    __shared__ float sv[1024];
    __shared__ int si[1024];
    for (int k = 0; k < K_TOP; k++) {
        float best = -3.0e38f; int bi = HWAC;
        for (int f = t; f < HWAC; f += 1024) {
            float v = s[f];
            if (v > best || (v == best && f < bi)) { best = v; bi = f; }
        }
        sv[t] = best; si[t] = bi;
        __syncthreads();
        for (int off = 512; off > 0; off >>= 1) {
            if (t < off) {
                float v2 = sv[t + off]; int i2 = si[t + off];
                if (v2 > sv[t] || (v2 == sv[t] && i2 < si[t])) { sv[t] = v2; si[t] = i2; }
            }
            __syncthreads();
        }
        if (t == 0) {
            topval[img * K_TOP + k] = sv[0];
            topidx[img * K_TOP + k] = si[0];
            s[si[0]] = -3.0e38f;   // mark so it is never re-picked
        }
        __threadfence_block();
        __syncthreads();
    }
}

// ---------------- Kernel 3: decode corners, WMMA projection, box2d + BEV ----------------
__global__ void decode_kernel(const float* __restrict__ reg,
                              const float* __restrict__ calib,
                              const int* __restrict__ img_sizes,
                              const float* __restrict__ anchors_y,
                              const int* __restrict__ topidx,
                              const float* __restrict__ topval,
                              float* __restrict__ out_box2d,
                              float* __restrict__ out_corners,
                              float* __restrict__ ws_bev,
                              int* __restrict__ ws_valid,
                              int* __restrict__ ws_cls) {
    __shared__ float pts[K_TOP * 8 * 3];   // 24 KB: 2048 corner points (x,y,z)
    __shared__ float uvw[K_TOP * 8 * 3];   // 24 KB: projected homogeneous coords
    int img = blockIdx.x;
    int b = threadIdx.x;
    int gid = img * K_TOP + b;
    int idx = topidx[gid];
    float score = topval[gid];
    int hw = idx / ACH;
    int ch = idx - hw * ACH;
    int ci = idx % NCLS;                    // cls_idx = top_idx % NUM_CLASSES
    int iz = hw / W_BEV;
    int ix = hw - iz * W_BEV;
    float lx = -30.0f + (ix + 0.5f) * 0.2f;
    float lz = 0.0f + (iz + 0.5f) * 0.2f;
    float ly = anchors_y[ci];
    const float* rbase = reg + (size_t)img * (ACH * REG_DIM) * HW + (size_t)(ch * REG_DIM) * HW + hw;
    for (int k = 0; k < 8; k++)
        for (int d = 0; d < 3; d++) {
            float rv = rbase[(size_t)(k * 3 + d) * HW];
            float add = (d == 0) ? lx : ((d == 1) ? ly : lz);
            pts[(b * 8 + k) * 3 + d] = rv + add;
        }
    __syncthreads();

    // --- WMMA f32 16x16x4: D(16 points x 16) = A(16x4 homog pts) x B(4x16 = P^T padded) ---
    const float* P = calib + img * 12;     // 3x4 row-major
    int lane = threadIdx.x & 31;
    int wv = threadIdx.x >> 5;
    int half = lane >> 4;                  // 0: K=0/1, 1: K=2/3 (and D rows M+8)
    int li = lane & 15;
    v2f bm; bm[0] = 0.0f; bm[1] = 0.0f;
    if (li < 3) {                          // column n=li holds P row li over K
        if (half == 0) { bm[0] = P[li * 4 + 0]; bm[1] = P[li * 4 + 1]; }
        else           { bm[0] = P[li * 4 + 2]; bm[1] = P[li * 4 + 3]; }
    }
    for (int tile = wv; tile < 128; tile += 8) {   // 128 tiles of 16 points
        int p = tile * 16 + li;
        // branchless operand build: no exec-mask dance in the WMMA feed
        float a0 = pts[p * 3 + (half << 1)];       // x (half=0) or z (half=1)
        float a1 = pts[p * 3 + 1];                 // y
        v2f am;
        am[0] = a0;
        am[1] = half ? 1.0f : a1;                  // K=3 is the homogeneous 1
        v8f cm = {};
        cm = __builtin_amdgcn_wmma_f32_16x16x4_f32(false, am, false, bm,
                                                   (short)0, cm, false, false);
        #pragma unroll
        for (int j = 0; j < 8; j++) {      // D: lane holds N=li, vgpr j -> M = j + half*8
            if (li < 3) uvw[(tile * 16 + j + half * 8) * 3 + li] = cm[j];
        }
    }
    __syncthreads();

    // --- per-box 2D box + clip ---
    float umin = 3.0e38f, vmin = 3.0e38f, umax = -3.0e38f, vmax = -3.0e38f;
    for (int k = 0; k < 8; k++) {
        float w = uvw[(b * 8 + k) * 3 + 2];
        float u = uvw[(b * 8 + k) * 3 + 0] / w;
        float v = uvw[(b * 8 + k) * 3 + 1] / w;
        umin = fminf(umin, u); umax = fmaxf(umax, u);
        vmin = fminf(vmin, v); vmax = fmaxf(vmax, v);
    }
    float himg = (float)img_sizes[img * 2 + 0];
    float wimg = (float)img_sizes[img * 2 + 1];
    float x1 = fminf(fmaxf(umin, 0.0f), wimg - 1.0f);
    float y1 = fminf(fmaxf(vmin, 0.0f), himg - 1.0f);
    float x2 = fminf(fmaxf(umax, 0.0f), wimg - 1.0f);
    float y2 = fminf(fmaxf(vmax, 0.0f), himg - 1.0f);
    out_box2d[gid * 4 + 0] = x1; out_box2d[gid * 4 + 1] = y1;
    out_box2d[gid * 4 + 2] = x2; out_box2d[gid * 4 + 3] = y2;
    int valid = (score > 0.0f) && (x2 - x1 >= 0.0f) && (y2 - y1 >= 0.0f);

    // --- BEV rotated box from 3D corners ---
    float cx = 0.0f, cz = 0.0f;
    for (int k = 0; k < 8; k++) { cx += pts[(b * 8 + k) * 3 + 0]; cz += pts[(b * 8 + k) * 3 + 2]; }
    cx *= 0.125f; cz *= 0.125f;
    float p0x = pts[(b * 8 + 0) * 3 + 0] - cx, p0z = pts[(b * 8 + 0) * 3 + 2] - cz;
    float p1x = pts[(b * 8 + 1) * 3 + 0] - cx, p1z = pts[(b * 8 + 1) * 3 + 2] - cz;
    float p2x = pts[(b * 8 + 2) * 3 + 0] - cx, p2z = pts[(b * 8 + 2) * 3 + 2] - cz;
    float elx = p0x - p1x, elz = p0z - p1z;
    float ewx = p1x - p2x, ewz = p1z - p2z;
    ws_bev[gid * 5 + 0] = cx;
    ws_bev[gid * 5 + 1] = cz;
    ws_bev[gid * 5 + 2] = sqrtf(ewx * ewx + ewz * ewz);
    ws_bev[gid * 5 + 3] = sqrtf(elx * elx + elz * elz);
    ws_bev[gid * 5 + 4] = atan2f(elz, elx);
    ws_valid[gid] = valid;
    ws_cls[gid] = ci;
    for (int k = 0; k < 8; k++)
        for (int d = 0; d < 3; d++)
            out_corners[gid * 24 + k * 3 + d] = pts[(b * 8 + k) * 3 + d];
}

// ---------------- Kernel 4: rotated-rect IoU matrix ----------------
__device__ __forceinline__ float cross2(float ax, float az, float bx, float bz,
                                        float px, float pz) {
    return (bx - ax) * (pz - az) - (bz - az) * (px - ax);
}

__device__ void rect_corners(const float* b, float* qx, float* qz) {
    float c = cosf(b[4]), s = sinf(b[4]);
    const float lxs[4] = {0.5f, -0.5f, -0.5f, 0.5f};
    const float wzs[4] = {0.5f, 0.5f, -0.5f, -0.5f};
    for (int i = 0; i < 4; i++) {
        float lx = lxs[i] * b[3], wz = wzs[i] * b[2];
        qx[i] = b[0] + c * lx - s * wz;
        qz[i] = b[1] + s * lx + c * wz;
    }
}

__device__ float pair_iou(const float* b1, const float* b2) {
    float q1x[4], q1z[4], q2x[4], q2z[4];
    rect_corners(b1, q1x, q1z);
    rect_corners(b2, q2x, q2z);
    float px[8], pz[8];
    int n = 4;
    for (int i = 0; i < 4; i++) { px[i] = q1x[i]; pz[i] = q1z[i]; }
    for (int e = 0; e < 4; e++) {
        float ax = q2x[e], az = q2z[e];
        float bx = q2x[(e + 1) & 3], bz = q2z[(e + 1) & 3];
        float nx[8], nz[8];
        int m = 0;
        for (int k = 0; k < n; k++) {
            int k2 = (k + 1 < n) ? k + 1 : 0;
            float dc = cross2(ax, az, bx, bz, px[k], pz[k]);
            float dn = cross2(ax, az, bx, bz, px[k2], pz[k2]);
            bool kc = dc >= 0.0f, kn = dn >= 0.0f;
            if (kc && m < 8) { nx[m] = px[k]; nz[m] = pz[k]; m++; }
            if (kc != kn && m < 8) {
                float den = dc - dn;
                if (den == 0.0f) den = 1.0f;
                float tt = dc / den;
                nx[m] = px[k] + tt * (px[k2] - px[k]);
                nz[m] = pz[k] + tt * (pz[k2] - pz[k]);
                m++;
            }
        }
        n = m;
        for (int k = 0; k < n; k++) { px[k] = nx[k]; pz[k] = nz[k]; }
    }
    float acc = 0.0f;
    for (int k = 0; k < n; k++) {
        int k2 = (k + 1 < n) ? k + 1 : 0;
        acc += px[k] * pz[k2] - px[k2] * pz[k];
    }
    float inter = 0.5f * fabsf(acc);
    float uni = b1[2] * b1[3] + b2[2] * b2[3] - inter;
    return inter / fmaxf(uni, 1e-8f);
}

__global__ void iou_kernel(const float* __restrict__ bev, float* __restrict__ iou) {
    int blk = blockIdx.x;          // img*256 + row
    int img = blk >> 8;
    int i = blk & 255;
    int j = threadIdx.x;
    float b1[5], b2[5];
    for (int d = 0; d < 5; d++) {
        b1[d] = bev[(size_t)(img * K_TOP + i) * 5 + d];
        b2[d] = bev[(size_t)(img * K_TOP + j) * 5 + d];
    }
    iou[(size_t)img * (K_TOP * K_TOP) + i * K_TOP + j] = pair_iou(b1, b2);
}

// ---------------- Kernel 5: greedy NMS + final threshold ----------------
__global__ void nms_kernel(const float* __restrict__ iou,
                           const float* __restrict__ topval,
                           const int* __restrict__ ws_valid,
                           const int* __restrict__ ws_cls,
                           float* __restrict__ out_scores,
                           float* __restrict__ out_labels,
                           float* __restrict__ out_final) {
    int img = blockIdx.x;
    int t = threadIdx.x;
    int gid = img * K_TOP + t;
    __shared__ int supp[K_TOP];
    __shared__ int keep[K_TOP];
    __shared__ float thr;
    float sc = topval[gid];
    int cls = ws_cls[gid];
    int val = ws_valid[gid];
    keep[t] = 0;
    __syncthreads();
    const float* iouM = iou + (size_t)img * (K_TOP * K_TOP);
    for (int j = 0; j < NCLS; j++) {
        supp[t] = !(val && cls == j);
        __syncthreads();
        for (int i = 0; i < K_TOP; i++) {
            int si = supp[i];               // supp[i] only mutated in earlier iterations
            if (si == 0 && t > i && iouM[i * K_TOP + t] > 0.1f) supp[t] = 1;
            __syncthreads();
        }
        if (!supp[t]) keep[t] = 1;
        __syncthreads();
    }
    if (t == 0) {
        int c = 0; float kv = -1.0f;
        for (int i = 0; i < K_TOP; i++) {   // topval is descending -> kept subset is too
            if (keep[i]) { if (c == 99) kv = topval[img * K_TOP + i]; c++; }
        }
        thr = (c > 100) ? fmaxf(kv, 0.05f) : 0.05f;
    }
    __syncthreads();
    out_scores[gid] = sc;
    out_labels[gid] = (float)(cls + 1);
    out_final[gid] = (keep[t] && sc >= thr) ? 1.0f : 0.0f;
}

// ---------------- Launch ----------------
extern "C" void kernel_launch(void* const* d_in, const int* in_sizes, int n_in,
                              void* d_out, int out_size, void* d_ws, size_t ws_size,
                              hipStream_t stream) {
    const float* box_cls   = (const float*)d_in[0];
    const float* box_reg   = (const float*)d_in[1];
    const float* ctr       = (const float*)d_in[2];
    const float* calib     = (const float*)d_in[3];
    const int*   img_sizes = (const int*)d_in[4];
    const float* anchors_y = (const float*)d_in[5];
    int N = in_sizes[0] / HWAC;   // box_cls = (N, 6, 350, 300)

    char* ws = (char*)d_ws;
    size_t off = 0;
    float* ws_scores = (float*)(ws + off); off += (size_t)N * HWAC * sizeof(float);
    int*   ws_topidx = (int*)(ws + off);   off += (size_t)N * K_TOP * sizeof(int);
    float* ws_topval = (float*)(ws + off); off += (size_t)N * K_TOP * sizeof(float);
    float* ws_bev    = (float*)(ws + off); off += (size_t)N * K_TOP * 5 * sizeof(float);
    int*   ws_valid  = (int*)(ws + off);   off += (size_t)N * K_TOP * sizeof(int);
    int*   ws_clsix  = (int*)(ws + off);   off += (size_t)N * K_TOP * sizeof(int);
    float* ws_iou    = (float*)(ws + off); off += (size_t)N * K_TOP * K_TOP * sizeof(float);

    float* out = (float*)d_out;
    float* out_box2d   = out;                                   // N*256*4
    float* out_scores  = out_box2d + (size_t)N * K_TOP * 4;     // N*256
    float* out_labels  = out_scores + (size_t)N * K_TOP;        // N*256
    float* out_corners = out_labels + (size_t)N * K_TOP;        // N*256*24
    float* out_final   = out_corners + (size_t)N * K_TOP * 24;  // N*256

    int total = N * HWAC;
    score_kernel<<<(total + 255) / 256, 256, 0, stream>>>(box_cls, ctr, ws_scores, total);
    topk_kernel<<<N, 1024, 0, stream>>>(ws_scores, ws_topidx, ws_topval);
    decode_kernel<<<N, 256, 0, stream>>>(box_reg, calib, img_sizes, anchors_y,
                                         ws_topidx, ws_topval, out_box2d, out_corners,
                                         ws_bev, ws_valid, ws_clsix);
    iou_kernel<<<N * K_TOP, 256, 0, stream>>>(ws_bev, ws_iou);
    nms_kernel<<<N, K_TOP, 0, stream>>>(ws_iou, ws_topval, ws_valid, ws_clsix,
                                        out_scores, out_labels, out_final);
}